// CortexBlock_79164837200261
// MI455X (gfx1250) — compile-verified
//
#include <hip/hip_runtime.h>
#include <hip/hip_bf16.h>
#include <stdint.h>

// ---------------- problem dims ----------------
#define DIM_B  2
#define DIM_S  1024
#define DIM_D  1024
#define DIM_H  16
#define DIM_HD 64
#define DIM_DI 2048
#define DIM_NS 4
#define DIM_M  256
#define DIM_E  8
#define DIM_FF 2048
#define BS     (DIM_B * DIM_S)   // 2048 rows
#define TOPK_K 128

typedef float v2f __attribute__((ext_vector_type(2)));
typedef float v8f __attribute__((ext_vector_type(8)));
typedef unsigned int u32x4 __attribute__((ext_vector_type(4)));
typedef int i32x4 __attribute__((ext_vector_type(4)));
typedef int i32x8 __attribute__((ext_vector_type(8)));

__device__ __forceinline__ v8f wmma4(v2f a, v2f b, v8f c) {
  // V_WMMA_F32_16X16X4_F32 : D = A(16x4) * B(4x16) + C(16x16)
  return __builtin_amdgcn_wmma_f32_16x16x4_f32(false, a, false, b, (short)0, c,
                                               false, false);
}

__device__ __forceinline__ float silu_f(float x) { return x / (1.f + __expf(-x)); }
__device__ __forceinline__ float sigm_f(float x) { return 1.f / (1.f + __expf(-x)); }
__device__ __forceinline__ unsigned fmap_u(float f) {
  unsigned u = __float_as_uint(f);
  return (u & 0x80000000u) ? ~u : (u | 0x80000000u);
}

// ---------------------------------------------------------------------------
// TDM: DMA a rows x cols f32 tile (row stride ldb elems) from global into LDS.
// D# packing per CDNA5 ISA ch.8 (group0: count/lds/global/type, group1: dims).
// ---------------------------------------------------------------------------
__device__ __forceinline__ void tdm_load_tile_f32(const float* gsrc,
                                                  unsigned lds_off, int ldb,
                                                  int tdim1, int rows, int cols) {
  const unsigned long long ga = (unsigned long long)(uintptr_t)gsrc;
  u32x4 g0;
  g0[0] = 1u;                                           // count=1 (user D#)
  g0[1] = lds_off;                                      // lds_addr
  g0[2] = (unsigned)(ga & 0xFFFFFFFFull);               // global_addr[31:0]
  g0[3] = (unsigned)((ga >> 32) & 0x1FFFFFFu) | (2u << 30);  // addr hi | type=2
  i32x8 g1;
  g1[0] = (int)(2u << 16);                              // data_size = 4B
  g1[1] = (int)(((unsigned)ldb & 0xFFFFu) << 16);       // tensor_dim0 lo16
  g1[2] = (int)((((unsigned)ldb >> 16) & 0xFFFFu) |
                (((unsigned)tdim1 & 0xFFFFu) << 16));   // dim0 hi | dim1 lo
  g1[3] = (int)((((unsigned)tdim1 >> 16) & 0xFFFFu) |
                (((unsigned)cols) << 16));              // dim1 hi | tile_dim0
  g1[4] = rows;                                         // tile_dim1 (tile_dim2=0)
  g1[5] = ldb;                                          // tensor_dim0_stride lo32
  g1[6] = 0;
  g1[7] = 0;
  i32x4 gz = {0, 0, 0, 0};
#if defined(__clang_major__) && (__clang_major__ >= 23)
  i32x8 gz8 = {0, 0, 0, 0, 0, 0, 0, 0};
  __builtin_amdgcn_tensor_load_to_lds(g0, g1, gz, gz, gz8, 0);
#else
  __builtin_amdgcn_tensor_load_to_lds(g0, g1, gz, gz, 0);
#endif
}

// ---------------------------------------------------------------------------
// f32 WMMA GEMM: C[M,N] (+)= A[M,K] @ B[K,N]; 64x128 block, 32x32 per wave.
// B tiles are TDM-DMA'd into double-buffered LDS (TENSORcnt pipelined);
// A tiles thread-staged with global prefetch of the next K strip.
// mode 0: store   1: store silu(acc)   2: C += rowscale[m]*acc   3: C += acc
// ---------------------------------------------------------------------------
__global__ __launch_bounds__(256) void gemm_f32_wmma(
    const float* __restrict__ A, const float* __restrict__ Bm,
    float* __restrict__ C, int M, int N, int K, int lda, int ldb, int ldc,
    int mode, const float* __restrict__ rowscale) {
  __shared__ float As[64][20];
  __shared__ float Bs[2][16][128];
  const int bm = blockIdx.y * 64, bn = blockIdx.x * 128;
  const int t = threadIdx.x, w = t >> 5, lane = t & 31;
  const int half = lane >> 4, ln = lane & 15;
  const int wm = (w >> 2) << 5;   // 0 / 32
  const int wn = (w & 3) << 5;    // 0/32/64/96
  v8f c00 = {0, 0, 0, 0, 0, 0, 0, 0};
  v8f c01 = {0, 0, 0, 0, 0, 0, 0, 0};
  v8f c10 = {0, 0, 0, 0, 0, 0, 0, 0};
  v8f c11 = {0, 0, 0, 0, 0, 0, 0, 0};

  const unsigned bs0 = (unsigned)(uintptr_t)(&Bs[0][0][0]);
  if (w == 0)  // preload B tile 0 via TDM
    tdm_load_tile_f32(Bm + bn, bs0, ldb, K, 16, 128);

  int buf = 0;
  for (int k0 = 0; k0 < K; k0 += 16) {
    { // stage A 64x16 (float4 per thread) + prefetch next strip
      const int li = t * 4;
      const int ra = li >> 4, ca = li & 15;
      const float4 va = *(const float4*)(A + (size_t)(bm + ra) * lda + k0 + ca);
      As[ra][ca + 0] = va.x; As[ra][ca + 1] = va.y;
      As[ra][ca + 2] = va.z; As[ra][ca + 3] = va.w;
      if (k0 + 16 < K)
        __builtin_prefetch(A + (size_t)(bm + (t >> 2)) * lda + k0 + 16, 0, 1);
    }
    if (w == 0) {  // pipeline: issue next B tile DMA, ensure current complete
      if (k0 + 16 < K) {
        tdm_load_tile_f32(Bm + (size_t)(k0 + 16) * ldb + bn,
                          bs0 + (unsigned)((buf ^ 1) * 16 * 128 * 4), ldb,
                          K - (k0 + 16), 16, 128);
        __builtin_amdgcn_s_wait_tensorcnt(1);
      } else {
        __builtin_amdgcn_s_wait_tensorcnt(0);
      }
    }
    __syncthreads();
#pragma unroll
    for (int kk = 0; kk < 16; kk += 4) {
      const int kb0 = kk + 2 * half;
      v2f a0; a0.x = As[wm + ln][kb0];       a0.y = As[wm + ln][kb0 + 1];
      v2f a1; a1.x = As[wm + 16 + ln][kb0];  a1.y = As[wm + 16 + ln][kb0 + 1];
      v2f b0; b0.x = Bs[buf][kb0][wn + ln];
              b0.y = Bs[buf][kb0 + 1][wn + ln];
      v2f b1; b1.x = Bs[buf][kb0][wn + 16 + ln];
              b1.y = Bs[buf][kb0 + 1][wn + 16 + ln];
      c00 = wmma4(a0, b0, c00);
      c01 = wmma4(a0, b1, c01);
      c10 = wmma4(a1, b0, c10);
      c11 = wmma4(a1, b1, c11);
    }
    __syncthreads();
    buf ^= 1;
  }
#pragma unroll
  for (int r = 0; r < 8; r++) {
    const int row0 = bm + wm + r + 8 * half;
    const int row1 = row0 + 16;
    const int col0 = bn + wn + ln;
    float* p00 = C + (size_t)row0 * ldc + col0;
    float* p10 = C + (size_t)row1 * ldc + col0;
    const float v00 = c00[r], v01 = c01[r], v10 = c10[r], v11 = c11[r];
    if (mode == 0) {
      p00[0] = v00; p00[16] = v01; p10[0] = v10; p10[16] = v11;
    } else if (mode == 1) {
      p00[0] = silu_f(v00); p00[16] = silu_f(v01);
      p10[0] = silu_f(v10); p10[16] = silu_f(v11);
    } else if (mode == 3) {
      p00[0] += v00; p00[16] += v01; p10[0] += v10; p10[16] += v11;
    } else {
      const float s0 = rowscale[row0], s1 = rowscale[row1];
      p00[0] += s0 * v00; p00[16] += s0 * v01;
      p10[0] += s1 * v10; p10[16] += s1 * v11;
    }
  }
}

// ---------------------------------------------------------------------------
// RMSNorm: one block per row of D=1024
// ---------------------------------------------------------------------------
__global__ __launch_bounds__(256) void rmsnorm_k(const float* __restrict__ x,
                                                 const float* __restrict__ wgt,
                                                 float* __restrict__ out) {
  const int row = blockIdx.x;
  const float* xr = x + (size_t)row * DIM_D;
  __shared__ float red[256];
  float ss = 0.f;
  for (int c = threadIdx.x; c < DIM_D; c += 256) { float v = xr[c]; ss += v * v; }
  red[threadIdx.x] = ss;
  __syncthreads();
  for (int s = 128; s > 0; s >>= 1) {
    if (threadIdx.x < s) red[threadIdx.x] += red[threadIdx.x + s];
    __syncthreads();
  }
  const float r = rsqrtf(red[0] / (float)DIM_D + 1e-6f);
  for (int c = threadIdx.x; c < DIM_D; c += 256)
    out[(size_t)row * DIM_D + c] = xr[c] * r * wgt[c];
}

// ---------------------------------------------------------------------------
// SSM selective scan: thread per (b, channel), 4 decay states, fused silu(z)
// ---------------------------------------------------------------------------
__global__ __launch_bounds__(256) void ssm_scan_k(const float* __restrict__ xz,
                                                  const float* __restrict__ logdec,
                                                  const float* __restrict__ scale,
                                                  float* __restrict__ ssm_y) {
  const int idx = blockIdx.x * 256 + threadIdx.x;  // B*DI
  const int b = idx / DIM_DI, d = idx % DIM_DI;
  float a[DIM_NS], h[DIM_NS], sc[DIM_NS];
#pragma unroll
  for (int n = 0; n < DIM_NS; n++) {
    a[n] = sigm_f(logdec[n * DIM_DI + d]);
    sc[n] = scale[n];
    h[n] = 0.f;
  }
  const float* base = xz + (size_t)b * DIM_S * (2 * DIM_DI) + d;
  float* yb = ssm_y + (size_t)b * DIM_S * DIM_DI + d;
  for (int tt = 0; tt < DIM_S; tt++) {
    const float xin = base[(size_t)tt * (2 * DIM_DI)];
    const float z   = base[(size_t)tt * (2 * DIM_DI) + DIM_DI];
    float y = 0.f;
#pragma unroll
    for (int n = 0; n < DIM_NS; n++) {
      h[n] = a[n] * h[n] + (1.f - a[n]) * xin;
      y += sc[n] * h[n];
    }
    yb[(size_t)tt * DIM_DI] = y * silu_f(z);
  }
}

// ---------------------------------------------------------------------------
// RoPE in-place on q and k ([B,S,D] layout, per-head pairs)
// ---------------------------------------------------------------------------
__global__ __launch_bounds__(256) void rope_k(float* __restrict__ q,
                                              float* __restrict__ k) {
  const int idx = blockIdx.x * 256 + threadIdx.x;  // B*S*D/2
  const int pairs = DIM_D / 2;
  const int row = idx / pairs;
  const int p = idx % pairs;
  const int s = row % DIM_S;
  const int hd = p % (DIM_HD / 2);
  const int h = p / (DIM_HD / 2);
  const float inv = __powf(10000.f, -2.f * (float)hd / (float)DIM_HD);
  const float ang = (float)s * inv;
  const float cs = __cosf(ang), sn = __sinf(ang);
  const size_t o = (size_t)row * DIM_D + h * DIM_HD + 2 * hd;
  float q1 = q[o], q2 = q[o + 1];
  q[o] = q1 * cs - q2 * sn; q[o + 1] = q1 * sn + q2 * cs;
  float k1 = k[o], k2 = k[o + 1];
  k[o] = k1 * cs - k2 * sn; k[o + 1] = k1 * sn + k2 * cs;
}

// ---------------------------------------------------------------------------
// Sparse (top-128) causal attention: one workgroup per (b,h,16-query tile)
// ---------------------------------------------------------------------------
#define SCROW 1028
__global__ __launch_bounds__(256) void attn_k(const float* __restrict__ q,
                                              const float* __restrict__ k,
                                              const float* __restrict__ v,
                                              float* __restrict__ attnb) {
  __shared__ float Sc[16][SCROW];
  __shared__ float Ql[16][68];
  __shared__ float Kl[8][16][68];
  __shared__ float Oacc[2][16][68];
  const int wg = blockIdx.x;
  const int qt = wg % (DIM_S / 16);
  const int bh = wg / (DIM_S / 16);
  const int h = bh % DIM_H, b = bh / DIM_H;
  const int q0 = qt * 16;
  const int t = threadIdx.x, w = t >> 5, lane = t & 31;
  const int half = lane >> 4, ln = lane & 15;

  for (int i = t; i < 16 * SCROW; i += 256) ((float*)Sc)[i] = -1.0e9f;
  for (int li = t; li < 16 * 64; li += 256) {
    int r = li >> 6, c = li & 63;
    Ql[r][c] = q[(size_t)(b * DIM_S + q0 + r) * DIM_D + h * DIM_HD + c];
  }
  __syncthreads();

  // ---- phase 1: scores = q @ k^T / 8, causal mask ----
  const float* kb = k + (size_t)(b * DIM_S) * DIM_D + h * DIM_HD;
  const int ntiles = qt + 1;
  for (int kt = w; kt < ntiles; kt += 8) {
    const int key0 = kt * 16;
    // per-wave coalesced staging of this 16-key tile
    for (int li = lane; li < 16 * 16; li += 32) {
      const int r = li >> 4, c4 = (li & 15) * 4;
      const float4 kv = *(const float4*)(kb + (size_t)(key0 + r) * DIM_D + c4);
      Kl[w][r][c4 + 0] = kv.x; Kl[w][r][c4 + 1] = kv.y;
      Kl[w][r][c4 + 2] = kv.z; Kl[w][r][c4 + 3] = kv.w;
    }
    v8f acc = {0, 0, 0, 0, 0, 0, 0, 0};
#pragma unroll
    for (int kk = 0; kk < DIM_HD; kk += 4) {
      v2f a;  a.x = Ql[ln][kk + 2 * half];    a.y = Ql[ln][kk + 2 * half + 1];
      v2f bb; bb.x = Kl[w][ln][kk + 2 * half]; bb.y = Kl[w][ln][kk + 2 * half + 1];
      acc = wmma4(a, bb, acc);
    }
#pragma unroll
    for (int r = 0; r < 8; r++) {
      const int m = r + 8 * half;
      const int key = key0 + ln;
      float sv = acc[r] * 0.125f;
      if (key > q0 + m) sv = -1.0e9f;
      Sc[m][key] = sv;
    }
  }
  __syncthreads();

  // ---- phase 2: exact top-128 threshold (bisection on order-mapped u32) ----
#pragma unroll 1
  for (int rr = 0; rr < 2; rr++) {
    const int m = w * 2 + rr;
    float vals[32]; unsigned mv[32];
#pragma unroll
    for (int j = 0; j < 32; j++) {
      float f = Sc[m][lane + 32 * j];
      vals[j] = f;
      mv[j] = fmap_u(f);
    }
    unsigned long long lo = 0ull, hi = 0x100000000ull;
    while (hi - lo > 1ull) {
      const unsigned midu = (unsigned)((lo + hi) >> 1);
      int cnt = 0;
#pragma unroll
      for (int j = 0; j < 32; j++) cnt += (mv[j] >= midu) ? 1 : 0;
      for (int off = 16; off; off >>= 1) cnt += __shfl_down(cnt, off, 32);
      cnt = __shfl(cnt, 0, 32);
      if (cnt >= TOPK_K) lo = (lo + hi) >> 1; else hi = (lo + hi) >> 1;
    }
    const unsigned thr = (unsigned)lo;
    float mx = -3.0e38f;
#pragma unroll
    for (int j = 0; j < 32; j++) mx = fmaxf(mx, vals[j]);
    for (int off = 16; off; off >>= 1) mx = fmaxf(mx, __shfl_down(mx, off, 32));
    mx = __shfl(mx, 0, 32);
    float sum = 0.f;
#pragma unroll
    for (int j = 0; j < 32; j++)
      sum += (mv[j] >= thr) ? __expf(vals[j] - mx) : 0.f;
    for (int off = 16; off; off >>= 1) sum += __shfl_down(sum, off, 32);
    sum = __shfl(sum, 0, 32);
    const float inv = 1.f / sum;
#pragma unroll
    for (int j = 0; j < 32; j++)
      Sc[m][lane + 32 * j] = (mv[j] >= thr) ? __expf(vals[j] - mx) * inv : 0.f;
  }
  __syncthreads();

  // ---- phase 3: out = probs @ v  (waves split: 4 d-tiles x 2 key-phases) ----
  const int ct = w & 3, kh = w >> 2, d0 = ct * 16;
  const float* vb = v + (size_t)(b * DIM_S) * DIM_D + h * DIM_HD;
  v8f o = {0, 0, 0, 0, 0, 0, 0, 0};
  const int limit = q0 + 16;
  for (int k0 = kh * 4; k0 < limit; k0 += 8) {
    v2f a;  a.x = Sc[ln][k0 + 2 * half]; a.y = Sc[ln][k0 + 2 * half + 1];
    const float* vr = vb + (size_t)(k0 + 2 * half) * DIM_D + d0 + ln;
    v2f bb; bb.x = vr[0]; bb.y = vr[DIM_D];
    o = wmma4(a, bb, o);
  }
#pragma unroll
  for (int r = 0; r < 8; r++) Oacc[kh][r + 8 * half][d0 + ln] = o[r];
  __syncthreads();
  for (int li = t; li < 16 * 64; li += 256) {
    const int m = li >> 6, c = li & 63;
    attnb[(size_t)(b * DIM_S + q0 + m) * DIM_D + h * DIM_HD + c] =
        Oacc[0][m][c] + Oacc[1][m][c];
  }
}

// ---------------------------------------------------------------------------
// Memory-matrix scan: 1 block per batch; 256x256 state in registers.
// ---------------------------------------------------------------------------
__global__ __launch_bounds__(1024) void memscan_k(const float* __restrict__ mq,
                                                  const float* __restrict__ mk,
                                                  const float* __restrict__ mv,
                                                  const float* __restrict__ dlogit,
                                                  float* __restrict__ r_out) {
  const int b = blockIdx.x;
  const int t = threadIdx.x;
  const int n = t & 255, g = t >> 8;  // g in 0..3 (64 m-rows each)
  float Mreg[64];
#pragma unroll
  for (int i = 0; i < 64; i++) Mreg[i] = 0.f;
  __shared__ float kq[512];      // k[0:256], q[256:512]
  __shared__ float red[4][256];
  const float decay = sigm_f(dlogit[0]);
  for (int step = 0; step < DIM_S; step++) {
    const size_t ro = (size_t)(b * DIM_S + step) * DIM_M;
    if (t < 256) kq[t] = mk[ro + t];
    else if (t < 512) kq[t] = mq[ro + (t - 256)];
    __syncthreads();
    const float vn = mv[ro + n];
    float part = 0.f;
#pragma unroll
    for (int i = 0; i < 64; i++) {
      const int m = g * 64 + i;
      const float mm = decay * Mreg[i] + kq[m] * vn;
      Mreg[i] = mm;
      part += kq[256 + m] * mm;
    }
    red[g][n] = part;
    __syncthreads();
    if (g == 0) r_out[ro + n] = red[0][n] + red[1][n] + red[2][n] + red[3][n];
    __syncthreads();
  }
}

// ---------------------------------------------------------------------------
// Fuse gate: silu(g1+b1) @ fuse_w2 + b2 -> softmax3 -> combine -> x1, out=x1
// ---------------------------------------------------------------------------
__global__ __launch_bounds__(256) void fuse_k(
    const float* __restrict__ g1raw, const float* __restrict__ b1,
    const float* __restrict__ w2, const float* __restrict__ b2,
    const float* __restrict__ ssm_o, const float* __restrict__ attn_o,
    const float* __restrict__ mem_o, const float* __restrict__ x,
    float* __restrict__ x1, float* __restrict__ outb) {
  const int row = blockIdx.x;
  __shared__ float red3[3][256];
  __shared__ float gw[3];
  const int t = threadIdx.x;
  const float hq = silu_f(g1raw[(size_t)row * 256 + t] + b1[t]);
  red3[0][t] = hq * w2[t * 3 + 0];
  red3[1][t] = hq * w2[t * 3 + 1];
  red3[2][t] = hq * w2[t * 3 + 2];
  __syncthreads();
  for (int s = 128; s > 0; s >>= 1) {
    if (t < s) {
      red3[0][t] += red3[0][t + s];
      red3[1][t] += red3[1][t + s];
      red3[2][t] += red3[2][t + s];
    }
    __syncthreads();
  }
  if (t == 0) {
    float g0 = red3[0][0] + b2[0], g1 = red3[1][0] + b2[1], g2 = red3[2][0] + b2[2];
    float mx = fmaxf(g0, fmaxf(g1, g2));
    float e0 = __expf(g0 - mx), e1 = __expf(g1 - mx), e2 = __expf(g2 - mx);
    float inv = 1.f / (e0 + e1 + e2);
    gw[0] = e0 * inv; gw[1] = e1 * inv; gw[2] = e2 * inv;
  }
  __syncthreads();
  const float w0 = gw[0], w1 = gw[1], w2v = gw[2];
  for (int c = t; c < DIM_D; c += 256) {
    const size_t o = (size_t)row * DIM_D + c;
    const float f = ssm_o[o] * w0 + attn_o[o] * w1 + mem_o[o] * w2v;
    const float xv = x[o] + f;
    x1[o] = xv;
    outb[o] = xv;
  }
}

// ---------------------------------------------------------------------------
// Router: probs = softmax(xn2 @ router_w); top-2 renormalized -> wtsT[E][BS]
// ---------------------------------------------------------------------------
__global__ __launch_bounds__(256) void router_k(const float* __restrict__ xn2,
                                                const float* __restrict__ rw,
                                                float* __restrict__ wtsT) {
  const int row = blockIdx.x;
  __shared__ float red[256];
  __shared__ float logits[DIM_E];
  float local[DIM_E];
#pragma unroll
  for (int e = 0; e < DIM_E; e++) local[e] = 0.f;
  for (int c = threadIdx.x; c < DIM_D; c += 256) {
    const float xv = xn2[(size_t)row * DIM_D + c];
#pragma unroll
    for (int e = 0; e < DIM_E; e++) local[e] += xv * rw[c * DIM_E + e];
  }
  for (int e = 0; e < DIM_E; e++) {
    red[threadIdx.x] = local[e];
    __syncthreads();
    for (int s = 128; s > 0; s >>= 1) {
      if (threadIdx.x < s) red[threadIdx.x] += red[threadIdx.x + s];
      __syncthreads();
    }
    if (threadIdx.x == 0) logits[e] = red[0];
    __syncthreads();
  }
  if (threadIdx.x == 0) {
    float mx = -3.0e38f, p[DIM_E];
#pragma unroll
    for (int e = 0; e < DIM_E; e++) mx = fmaxf(mx, logits[e]);
    float se = 0.f;
#pragma unroll
    for (int e = 0; e < DIM_E; e++) { p[e] = __expf(logits[e] - mx); se += p[e]; }
#pragma unroll
    for (int e = 0; e < DIM_E; e++) p[e] /= se;
    float m1 = -1.f, m2 = -1.f;
#pragma unroll
    for (int e = 0; e < DIM_E; e++) {
      if (p[e] > m1) { m2 = m1; m1 = p[e]; }
      else if (p[e] > m2) m2 = p[e];
    }
    float ssum = 0.f, wv[DIM_E];
#pragma unroll
    for (int e = 0; e < DIM_E; e++) { wv[e] = (p[e] >= m2) ? p[e] : 0.f; ssum += wv[e]; }
    const float inv = 1.f / ssum;
#pragma unroll
    for (int e = 0; e < DIM_E; e++) wtsT[(size_t)e * BS + row] = wv[e] * inv;
  }
}

// ---------------------------------------------------------------------------
static inline void gemm(hipStream_t st, const float* A, const float* B,
                        float* C, int M, int N, int K, int lda, int ldb,
                        int ldc, int mode, const float* rs) {
  dim3 g(N / 128, M / 64);
  gemm_f32_wmma<<<g, 256, 0, st>>>(A, B, C, M, N, K, lda, ldb, ldc, mode, rs);
}

extern "C" void kernel_launch(void* const* d_in, const int* in_sizes, int n_in,
                              void* d_out, int out_size, void* d_ws,
                              size_t ws_size, hipStream_t stream) {
  (void)in_sizes; (void)n_in; (void)out_size; (void)ws_size;
  const float* x        = (const float*)d_in[0];
  const float* norm1_w  = (const float*)d_in[1];
  const float* norm2_w  = (const float*)d_in[2];
  const float* ssm_in_w = (const float*)d_in[3];
  const float* ssm_out_w= (const float*)d_in[4];
  const float* ssm_ld   = (const float*)d_in[5];
  const float* ssm_sc   = (const float*)d_in[6];
  const float* wq       = (const float*)d_in[7];
  const float* wk       = (const float*)d_in[8];
  const float* wv       = (const float*)d_in[9];
  const float* wo       = (const float*)d_in[10];
  const float* mem_wq   = (const float*)d_in[11];
  const float* mem_wk   = (const float*)d_in[12];
  const float* mem_wv   = (const float*)d_in[13];
  const float* mem_wo   = (const float*)d_in[14];
  const float* mem_dl   = (const float*)d_in[15];
  const float* fuse_w1  = (const float*)d_in[16];
  const float* fuse_b1  = (const float*)d_in[17];
  const float* fuse_w2  = (const float*)d_in[18];
  const float* fuse_b2  = (const float*)d_in[19];
  const float* router_w = (const float*)d_in[20];
  const float* moe_w1   = (const float*)d_in[21];
  const float* moe_w2   = (const float*)d_in[22];
  float* out = (float*)d_out;

  float* ws = (float*)d_ws;
  size_t o = 0;
  float* xn      = ws + o; o += (size_t)BS * DIM_D;
  float* xz      = ws + o; o += (size_t)BS * 2 * DIM_DI;
  float* ssm_y   = ws + o; o += (size_t)BS * DIM_DI;
  float* ssm_out = ws + o; o += (size_t)BS * DIM_D;
  float* qb      = ws + o; o += (size_t)BS * DIM_D;
  float* kb      = ws + o; o += (size_t)BS * DIM_D;
  float* vb      = ws + o; o += (size_t)BS * DIM_D;
  float* attnb   = ws + o; o += (size_t)BS * DIM_D;
  float* attn_o  = ws + o; o += (size_t)BS * DIM_D;
  float* mqb     = ws + o; o += (size_t)BS * DIM_M;
  float* mkb     = ws + o; o += (size_t)BS * DIM_M;
  float* mvb     = ws + o; o += (size_t)BS * DIM_M;
  float* mrb     = ws + o; o += (size_t)BS * DIM_M;
  float* mem_o   = ws + o; o += (size_t)BS * DIM_D;
  float* g1raw   = ws + o; o += (size_t)BS * 256;
  float* x1      = ws + o; o += (size_t)BS * DIM_D;
  float* xn2     = ws + o; o += (size_t)BS * DIM_D;
  float* wtsT    = ws + o; o += (size_t)DIM_E * BS;
  float* hmoe    = ws + o; o += (size_t)BS * DIM_FF;

  // 1) xn = rmsnorm(x)
  rmsnorm_k<<<BS, 256, 0, stream>>>(x, norm1_w, xn);
  // 2) xz = xn @ ssm_in_w
  gemm(stream, xn, ssm_in_w, xz, BS, 2 * DIM_DI, DIM_D, DIM_D, 2 * DIM_DI,
       2 * DIM_DI, 0, nullptr);
  // 3) SSM scan fused with silu(z)
  ssm_scan_k<<<(DIM_B * DIM_DI) / 256, 256, 0, stream>>>(xz, ssm_ld, ssm_sc, ssm_y);
  // 4) ssm_out = ssm_y @ ssm_out_w
  gemm(stream, ssm_y, ssm_out_w, ssm_out, BS, DIM_D, DIM_DI, DIM_DI, DIM_D,
       DIM_D, 0, nullptr);
  // 5) q,k,v projections
  gemm(stream, xn, wq, qb, BS, DIM_D, DIM_D, DIM_D, DIM_D, DIM_D, 0, nullptr);
  gemm(stream, xn, wk, kb, BS, DIM_D, DIM_D, DIM_D, DIM_D, DIM_D, 0, nullptr);
  gemm(stream, xn, wv, vb, BS, DIM_D, DIM_D, DIM_D, DIM_D, DIM_D, 0, nullptr);
  // 6) RoPE
  rope_k<<<(BS * DIM_D / 2) / 256, 256, 0, stream>>>(qb, kb);
  // 7) top-k causal attention
  attn_k<<<DIM_B * DIM_H * (DIM_S / 16), 256, 0, stream>>>(qb, kb, vb, attnb);
  // 8) attn_out = attnb @ wo
  gemm(stream, attnb, wo, attn_o, BS, DIM_D, DIM_D, DIM_D, DIM_D, DIM_D, 0, nullptr);
  // 9) memory-matrix projections
  gemm(stream, xn, mem_wq, mqb, BS, DIM_M, DIM_D, DIM_D, DIM_M, DIM_M, 0, nullptr);
  gemm(stream, xn, mem_wk, mkb, BS, DIM_M, DIM_D, DIM_D, DIM_M, DIM_M, 0, nullptr);
  gemm(stream, xn, mem_wv, mvb, BS, DIM_M, DIM_D, DIM_D, DIM_M, DIM_M, 0, nullptr);
  // 10) sequential memory scan
  memscan_k<<<DIM_B, 1024, 0, stream>>>(mqb, mkb, mvb, mem_dl, mrb);
  // 11) mem_out = r @ mem_wo
  gemm(stream, mrb, mem_wo, mem_o, BS, DIM_D, DIM_M, DIM_M, DIM_D, DIM_D, 0, nullptr);
  // 12) fuse pre-activation: g1raw = [ssm|attn|mem] @ fuse_w1 (row slices)
  gemm(stream, ssm_out, fuse_w1,                           g1raw, BS, 256,
       DIM_D, DIM_D, 256, 256, 0, nullptr);
  gemm(stream, attn_o,  fuse_w1 + (size_t)DIM_D * 256,     g1raw, BS, 256,
       DIM_D, DIM_D, 256, 256, 3, nullptr);
  gemm(stream, mem_o,   fuse_w1 + (size_t)2 * DIM_D * 256, g1raw, BS, 256,
       DIM_D, DIM_D, 256, 256, 3, nullptr);
  // 13) gate softmax + combine + residual; initializes d_out = x1
  fuse_k<<<BS, 256, 0, stream>>>(g1raw, fuse_b1, fuse_w2, fuse_b2, ssm_out,
                                 attn_o, mem_o, x, x1, out);
  // 14) xn2 = rmsnorm(x1)
  rmsnorm_k<<<BS, 256, 0, stream>>>(x1, norm2_w, xn2);
  // 15) router weights (top-2, renormalized)
  router_k<<<BS, 256, 0, stream>>>(xn2, router_w, wtsT);
  // 16) MoE: d_out += wts_e[row] * (silu(xn2 @ W1_e) @ W2_e)
  for (int e = 0; e < DIM_E; e++) {
    gemm(stream, xn2, moe_w1 + (size_t)e * DIM_D * DIM_FF, hmoe, BS, DIM_FF,
         DIM_D, DIM_D, DIM_FF, DIM_FF, 1, nullptr);
    gemm(stream, hmoe, moe_w2 + (size_t)e * DIM_FF * DIM_D, out, BS, DIM_D,
         DIM_FF, DIM_FF, DIM_D, DIM_D, 2, wtsT + (size_t)e * BS);
  }
}